// ActorCritic_45612552684141
// MI455X (gfx1250) — compile-verified
//
#include <hip/hip_runtime.h>
#include <math.h>

typedef __attribute__((ext_vector_type(2))) float v2f;
typedef __attribute__((ext_vector_type(8))) float v8f;
typedef __attribute__((ext_vector_type(4))) int   v4i;

#define NODE_NUM 19
#define N_EDGES  36

#if defined(__has_builtin)
#if __has_builtin(__builtin_amdgcn_global_load_async_to_lds_b128) && \
    __has_builtin(__builtin_amdgcn_s_wait_asynccnt)
#define HAVE_ASYNC_LDS 1
#endif
#endif

// ---------------------------------------------------------------------------
// Generic full-precision GEMV via V_WMMA_F32_16X16X4_F32.
//   y[0..out_f) = act( W(out_f x in_f) @ x(in_f) + bias ), in_f % 4 == 0,
//   in_f <= MAXK.
// One 16-row tile per block; K split across NWAVES waves; LDS reduction.
// x is staged into LDS with GLOBAL_LOAD_ASYNC_TO_LDS_B128 (ASYNCcnt) so the
// VMEM pipe is dedicated to streaming W; B operand then comes from LDS.
// A tile: lane(0..15)=rows, lane halves carry K{0,1}/K{2,3} per ISA layout.
// B tile: x[k] broadcast across all 16 columns -> every column of D equal.
// ---------------------------------------------------------------------------
template <int NWAVES, int MAXK>
__global__ __launch_bounds__(NWAVES * 32)
void gemv_wmma_k(const float* __restrict__ W, const float* __restrict__ bias,
                 const float* __restrict__ x, float* __restrict__ y,
                 int in_f, int out_f, int do_relu)
{
    __shared__ __align__(16) float xs[MAXK];
    __shared__ float red[NWAVES][32][8];

    const int lane = threadIdx.x & 31;
    const int wave = threadIdx.x >> 5;
    const int tile = blockIdx.x;
    const int row  = tile * 16 + (lane & 15);
    const int hi   = (lane >= 16) ? 2 : 0;         // lane half selects K pair
    const bool rowok = row < out_f;
    const float* wrow = W + (long long)row * in_f;

    // ---- stage x into LDS (bulk async copy, no VGPR data movement) ----
#if HAVE_ASYNC_LDS
    {
        typedef __attribute__((address_space(1))) v4i* gp_t;   // global int4*
        typedef __attribute__((address_space(3))) v4i* lp_t;   // LDS int4*
        for (int i = threadIdx.x; i < (in_f >> 2); i += NWAVES * 32) {
            __builtin_amdgcn_global_load_async_to_lds_b128(
                (gp_t)(x + 4 * i), (lp_t)(xs + 4 * i), 0, 0);
        }
        __builtin_amdgcn_s_wait_asynccnt(0);
    }
#else
    for (int i = threadIdx.x; i < in_f; i += NWAVES * 32) xs[i] = x[i];
#endif
    __syncthreads();

    v8f acc0 = {0.f, 0.f, 0.f, 0.f, 0.f, 0.f, 0.f, 0.f};
    v8f acc1 = acc0, acc2 = acc0, acc3 = acc0;
    const int ngroups = in_f >> 2;                 // K groups of 4
    int g = wave;

    // 4 independent accumulation chains: keep XDL pipe and W-stream loads busy
    for (; g + 3 * NWAVES < ngroups; g += 4 * NWAVES) {
        int k0 = (g + 0 * NWAVES) * 4 + hi;
        int k1 = (g + 1 * NWAVES) * 4 + hi;
        int k2 = (g + 2 * NWAVES) * 4 + hi;
        int k3 = (g + 3 * NWAVES) * 4 + hi;
        v2f a0 = {0.f, 0.f}, a1 = a0, a2 = a0, a3 = a0;
        if (rowok) {
            a0 = *(const v2f*)(wrow + k0);
            a1 = *(const v2f*)(wrow + k1);
            a2 = *(const v2f*)(wrow + k2);
            a3 = *(const v2f*)(wrow + k3);
        }
        if (in_f >= 2048) __builtin_prefetch(wrow + k0 + 2048, 0, 0);
        v2f b0 = *(const v2f*)(xs + k0);
        v2f b1 = *(const v2f*)(xs + k1);
        v2f b2 = *(const v2f*)(xs + k2);
        v2f b3 = *(const v2f*)(xs + k3);
        acc0 = __builtin_amdgcn_wmma_f32_16x16x4_f32(false, a0, false, b0, (short)0, acc0, false, false);
        acc1 = __builtin_amdgcn_wmma_f32_16x16x4_f32(false, a1, false, b1, (short)0, acc1, false, false);
        acc2 = __builtin_amdgcn_wmma_f32_16x16x4_f32(false, a2, false, b2, (short)0, acc2, false, false);
        acc3 = __builtin_amdgcn_wmma_f32_16x16x4_f32(false, a3, false, b3, (short)0, acc3, false, false);
    }
    for (; g < ngroups; g += NWAVES) {
        int k = g * 4 + hi;
        v2f a = {0.f, 0.f};
        if (rowok) a = *(const v2f*)(wrow + k);
        v2f b = *(const v2f*)(xs + k);
        acc0 = __builtin_amdgcn_wmma_f32_16x16x4_f32(false, a, false, b, (short)0, acc0, false, false);
    }

#pragma unroll
    for (int r = 0; r < 8; ++r)
        red[wave][lane][r] = (acc0[r] + acc1[r]) + (acc2[r] + acc3[r]);
    __syncthreads();

    // D layout: vgpr r, lanes 0-15 -> M=r (N=lane); lanes 16-31 -> M=r+8.
    // Column N=0 lives in lane 0 (M<8) / lane 16 (M>=8).
    if (threadIdx.x < 16) {
        const int M = threadIdx.x;
        const int srcLane = (M >= 8) ? 16 : 0;
        const int r = M & 7;
        float s = 0.f;
#pragma unroll
        for (int w = 0; w < NWAVES; ++w) s += red[w][srcLane][r];
        const int orow = tile * 16 + M;
        if (orow < out_f) {
            float val = s + bias[orow];
            if (do_relu) val = fmaxf(val, 0.f);
            y[orow] = val;
        }
    }
}

// ---------------------------------------------------------------------------
// conv1 (5x5 s2 p1) + BN + ReLU + 2x2 maxpool  : (1,125,125) -> (16,31,31)
// ---------------------------------------------------------------------------
__global__ __launch_bounds__(128)
void conv1_pool_k(const float* __restrict__ x, const float* __restrict__ w,
                  const float* __restrict__ cb,
                  const float* __restrict__ g, const float* __restrict__ bb,
                  const float* __restrict__ m, const float* __restrict__ vv,
                  float* __restrict__ out)
{
    int idx = blockIdx.x * blockDim.x + threadIdx.x;
    if (idx >= 16 * 31 * 31) return;
    int c = idx / 961;
    int rem = idx - c * 961;
    int py = rem / 31, px = rem % 31;
    float inv  = g[c] / sqrtf(vv[c] + 1e-5f);
    float beta = bb[c] - m[c] * inv;
    const float* wk = w + c * 25;
    float best = -3.402823466e38f;
#pragma unroll
    for (int sy = 0; sy < 2; ++sy)
#pragma unroll
        for (int sx = 0; sx < 2; ++sx) {
            int oy = 2 * py + sy, ox = 2 * px + sx;     // conv out in [0,62)
            int iy0 = oy * 2 - 1, ix0 = ox * 2 - 1;     // pad 1
            float acc = cb[c];
            for (int ky = 0; ky < 5; ++ky) {
                int iy = iy0 + ky;
                if ((unsigned)iy >= 125u) continue;
                for (int kx = 0; kx < 5; ++kx) {
                    int ix = ix0 + kx;
                    if ((unsigned)ix >= 125u) continue;
                    acc += x[iy * 125 + ix] * wk[ky * 5 + kx];
                }
            }
            best = fmaxf(best, fmaxf(acc * inv + beta, 0.f));
        }
    out[idx] = best;
}

// ---------------------------------------------------------------------------
// conv2 (3x3 s2) + BN + ReLU : (16,31,31) -> flat (32*15*15 = 7200), NCHW order
// ---------------------------------------------------------------------------
__global__ __launch_bounds__(128)
void conv2_k(const float* __restrict__ P, const float* __restrict__ w,
             const float* __restrict__ cb,
             const float* __restrict__ g, const float* __restrict__ bb,
             const float* __restrict__ m, const float* __restrict__ vv,
             float* __restrict__ out)
{
    int idx = blockIdx.x * blockDim.x + threadIdx.x;
    if (idx >= 32 * 15 * 15) return;
    int c = idx / 225;
    int rem = idx - c * 225;
    int y = rem / 15, xx = rem % 15;
    float acc = cb[c];
    const float* wc = w + c * 16 * 9;
    for (int ic = 0; ic < 16; ++ic) {
        const float* pi = P + ic * 961 + (y * 2) * 31 + xx * 2;
        const float* wi = wc + ic * 9;
#pragma unroll
        for (int ky = 0; ky < 3; ++ky)
#pragma unroll
            for (int kx = 0; kx < 3; ++kx)
                acc += pi[ky * 31 + kx] * wi[ky * 3 + kx];
    }
    float inv = g[c] / sqrtf(vv[c] + 1e-5f);
    out[idx] = fmaxf(acc * inv + (bb[c] - m[c] * inv), 0.f);
}

// ---------------------------------------------------------------------------
// Graph path: sage1 -> relu -> sage2 -> relu -> mean over nodes. 1 block/128 thr.
// edge_index stored (2,36) row-major: src = ei[0..35], dst = ei[36..71].
// ---------------------------------------------------------------------------
__global__ __launch_bounds__(128)
void graph_k(const float* __restrict__ xg, const int* __restrict__ ei,
             const float* __restrict__ wl1, const float* __restrict__ bl1,
             const float* __restrict__ wr1,
             const float* __restrict__ wl2, const float* __restrict__ bl2,
             const float* __restrict__ wr2,
             float* __restrict__ gmean)
{
    __shared__ float agg[NODE_NUM];
    __shared__ float g1[NODE_NUM][128];
    __shared__ float agg2[NODE_NUM][128];
    const int t = threadIdx.x;

    if (t < NODE_NUM) {
        float s = 0.f;
        for (int e = 0; e < N_EDGES; ++e)
            if (ei[N_EDGES + e] == t) s += xg[ei[e]];
        agg[t] = s;
    }
    __syncthreads();

    {   // sage1: feature t for every node (wl1, wr1 are (128,1))
        const float a = wl1[t], r = wr1[t], b = bl1[t];
        for (int n = 0; n < NODE_NUM; ++n)
            g1[n][t] = fmaxf(agg[n] * a + b + xg[n] * r, 0.f);
    }
    __syncthreads();

    {   // segment-sum of g1 over edges; thread t owns column t (no conflicts)
        for (int n = 0; n < NODE_NUM; ++n) agg2[n][t] = 0.f;
        for (int e = 0; e < N_EDGES; ++e)
            agg2[ei[N_EDGES + e]][t] += g1[ei[e]][t];
    }
    __syncthreads();

    {   // sage2 + relu + mean
        float sum = 0.f;
        for (int n = 0; n < NODE_NUM; ++n) {
            float acc = bl2[t];
            for (int c = 0; c < 128; ++c)
                acc += agg2[n][c] * wl2[t * 128 + c] + g1[n][c] * wr2[t * 128 + c];
            sum += fmaxf(acc, 0.f);
        }
        gmean[t] = sum * (1.0f / (float)NODE_NUM);
    }
}

// ---------------------------------------------------------------------------
// Head: actor logits + softmax + critic. One wave32.
// ---------------------------------------------------------------------------
__global__ __launch_bounds__(32)
void head_k(const float* __restrict__ feat,
            const float* __restrict__ aw, const float* __restrict__ ab,
            const float* __restrict__ cw, const float* __restrict__ cb,
            float* __restrict__ out)
{
    const int l = threadIdx.x;
    float logit = -3.402823466e38f;
    if (l < 19) {
        float a = ab[l];
        for (int c = 0; c < 128; ++c) a += aw[l * 128 + c] * feat[c];
        logit = a;
    }
    float mx = logit;
#pragma unroll
    for (int off = 16; off; off >>= 1) mx = fmaxf(mx, __shfl_xor(mx, off, 32));
    float e = (l < 19) ? __expf(logit - mx) : 0.f;
    float s = e;
#pragma unroll
    for (int off = 16; off; off >>= 1) s += __shfl_xor(s, off, 32);
    if (l < 19) out[l] = e / s;

    float cv = feat[l] * cw[l] + feat[l + 32] * cw[l + 32] +
               feat[l + 64] * cw[l + 64] + feat[l + 96] * cw[l + 96];
#pragma unroll
    for (int off = 16; off; off >>= 1) cv += __shfl_xor(cv, off, 32);
    if (l == 0) out[19] = cv + cb[0];
}

// ---------------------------------------------------------------------------
extern "C" void kernel_launch(void* const* d_in, const int* in_sizes, int n_in,
                              void* d_out, int out_size, void* d_ws, size_t ws_size,
                              hipStream_t stream)
{
    const float* x       = (const float*)d_in[0];
    const float* state   = (const float*)d_in[1];
    const float* x_graph = (const float*)d_in[2];
    const int*   edges   = (const int*)d_in[3];
    const float* c1w = (const float*)d_in[4];  const float* c1b = (const float*)d_in[5];
    const float* b1g = (const float*)d_in[6];  const float* b1b = (const float*)d_in[7];
    const float* b1m = (const float*)d_in[8];  const float* b1v = (const float*)d_in[9];
    const float* c2w = (const float*)d_in[10]; const float* c2b = (const float*)d_in[11];
    const float* b2g = (const float*)d_in[12]; const float* b2b = (const float*)d_in[13];
    const float* b2m = (const float*)d_in[14]; const float* b2v = (const float*)d_in[15];
    const float* fc0w = (const float*)d_in[16]; const float* fc0b = (const float*)d_in[17];
    const float* fc1w = (const float*)d_in[18]; const float* fc1b = (const float*)d_in[19];
    const float* fc2w = (const float*)d_in[20]; const float* fc2b = (const float*)d_in[21];
    const float* fc3w = (const float*)d_in[22]; const float* fc3b = (const float*)d_in[23];
    const float* s1wl = (const float*)d_in[24]; const float* s1bl = (const float*)d_in[25];
    const float* s1wr = (const float*)d_in[26];
    const float* s2wl = (const float*)d_in[27]; const float* s2bl = (const float*)d_in[28];
    const float* s2wr = (const float*)d_in[29];
    const float* fcgw = (const float*)d_in[30]; const float* fcgb = (const float*)d_in[31];
    const float* fc4w = (const float*)d_in[32]; const float* fc4b = (const float*)d_in[33];
    const float* aw   = (const float*)d_in[34]; const float* ab   = (const float*)d_in[35];
    const float* cw   = (const float*)d_in[36]; const float* cb   = (const float*)d_in[37];

    float* ws     = (float*)d_ws;          // ~95 KB of scratch used
    float* pooled = ws;                    // 16*31*31 = 15376
    float* flat   = ws + 15376;            // 7200
    float* h1     = ws + 22576;            // 512
    float* comb   = ws + 23088;            // 256 = [h2(128) | s2(64) | g(64)]
    float* h2     = comb;
    float* s2     = comb + 128;
    float* gout   = comb + 192;
    float* s1     = ws + 23344;            // 64
    float* gmean  = ws + 23408;            // 128
    float* feat   = ws + 23536;            // 128

    conv1_pool_k<<<121, 128, 0, stream>>>(x, c1w, c1b, b1g, b1b, b1m, b1v, pooled);
    conv2_k<<<57, 128, 0, stream>>>(pooled, c2w, c2b, b2g, b2b, b2m, b2v, flat);
    graph_k<<<1, 128, 0, stream>>>(x_graph, edges, s1wl, s1bl, s1wr,
                                   s2wl, s2bl, s2wr, gmean);

    // state path
    gemv_wmma_k<2, 64><<<4, 64, 0, stream>>>(fc2w, fc2b, state, s1, 4, 64, 1);
    gemv_wmma_k<2, 64><<<4, 64, 0, stream>>>(fc3w, fc3b, s1, s2, 64, 64, 1);
    // graph head
    gemv_wmma_k<2, 128><<<4, 64, 0, stream>>>(fcgw, fcgb, gmean, gout, 128, 64, 1);
    // conv head (dominant: 14.7 MB of fc0 weights streamed by 32 blocks x 8 waves)
    gemv_wmma_k<8, 7200><<<32, 256, 0, stream>>>(fc0w, fc0b, flat, h1, 7200, 512, 1);
    gemv_wmma_k<8, 512><<<8, 256, 0, stream>>>(fc1w, fc1b, h1, h2, 512, 128, 1);
    // combined
    gemv_wmma_k<2, 256><<<8, 64, 0, stream>>>(fc4w, fc4b, comb, feat, 256, 128, 1);
    head_k<<<1, 32, 0, stream>>>(feat, aw, ab, cw, cb, (float*)d_out);
}